// SemScalLoss_11768210391661
// MI455X (gfx1250) — compile-verified
//
#include <hip/hip_runtime.h>
#include <hip/hip_bf16.h>

typedef float v2f __attribute__((ext_vector_type(2)));
typedef float v8f __attribute__((ext_vector_type(8)));

#define C_NUM 18
#define C_PAD 32
#define BLOCK 256
#define BLOCKS_PER_B 256

__device__ __forceinline__ int imin(int a, int b) { return a < b ? a : b; }

// ---------------------------------------------------------------------------
// Main streaming kernel: one pass over pred/gt.
// Each block owns (b, chunk-of-n). Threads process 4 consecutive voxels per
// iteration via float4/int4 loads (fully coalesced b128 streams, 18 streams).
// p_sum accumulates in registers; nom / count go through LDS ds_add_f32.
// Block-level p_sum reduction (256x18 fp32 -> colsum) is done with
// V_WMMA_F32_16X16X4_F32 using an all-ones A matrix (exact fp32 math).
// ---------------------------------------------------------------------------
__global__ __launch_bounds__(BLOCK)
void semscal_main(const float* __restrict__ pred, const int* __restrict__ gt,
                  float* __restrict__ g_psum, float* __restrict__ g_nom,
                  float* __restrict__ g_cnt, int N) {
  __shared__ float tile[BLOCK * C_PAD];   // 32 KB: per-thread p_sum partials
  __shared__ float s_nom[C_NUM];
  __shared__ float s_cnt[C_NUM];

  const int tid = threadIdx.x;
  const int b   = blockIdx.x / BLOCKS_PER_B;
  const int j   = blockIdx.x % BLOCKS_PER_B;

  if (tid < C_NUM) { s_nom[tid] = 0.f; s_cnt[tid] = 0.f; }
  __syncthreads();

  const int G   = N >> 2;                               // groups of 4 voxels
  const int gpb = (G + BLOCKS_PER_B - 1) / BLOCKS_PER_B;
  const int q0  = j * gpb;
  const int q1  = imin(G, q0 + gpb);

  const float* predb = pred + (size_t)b * C_NUM * N;
  const int*   gtb   = gt   + (size_t)b * N;

  float acc[C_NUM];
#pragma unroll
  for (int c = 0; c < C_NUM; ++c) acc[c] = 0.f;

  for (int q = q0 + tid; q < q1; q += BLOCK) {
    const int n = q << 2;
    // prefetch next iteration's gt cacheline (gfx1250 global_prefetch_b8)
    __builtin_prefetch(gtb + n + (BLOCK << 2), 0, 1);

    float x[C_NUM][4];
#pragma unroll
    for (int c = 0; c < C_NUM; ++c) {
      float4 t = *(const float4*)(predb + (size_t)c * N + n);
      x[c][0] = t.x; x[c][1] = t.y; x[c][2] = t.z; x[c][3] = t.w;
    }
    int4 gi = *(const int4*)(gtb + n);
    int gv[4];
    gv[0] = gi.x; gv[1] = gi.y; gv[2] = gi.z; gv[3] = gi.w;

#pragma unroll
    for (int v = 0; v < 4; ++v) {
      float m = -3.402823466e38f;
#pragma unroll
      for (int c = 0; c < C_NUM; ++c) m = fmaxf(m, x[c][v]);

      float e[C_NUM];
      float Z = 0.f;
#pragma unroll
      for (int c = 0; c < C_NUM; ++c) {
        e[c] = __expf(x[c][v] - m);
        Z += e[c];
      }
      const float rZ = __builtin_amdgcn_rcpf(Z);

      const int gc = gv[v];
      float pg = 0.f;
#pragma unroll
      for (int c = 0; c < C_NUM; ++c) {
        float p = e[c] * rZ;
        acc[c] += p;
        pg = (c == gc) ? p : pg;              // v_cndmask select of p[gt]
      }
      atomicAdd(&s_nom[gc], pg);              // ds_add_f32
      atomicAdd(&s_cnt[gc], 1.0f);            // ds_add_f32
    }
  }

  // stage per-thread accumulators into LDS matrix [256 x 32], cols >=18 zero
#pragma unroll
  for (int c = 0; c < C_PAD; ++c)
    tile[tid * C_PAD + c] = (c < C_NUM) ? acc[c] : 0.f;
  __syncthreads();

  // Wave 0: column-sum of tile via V_WMMA_F32_16X16X4_F32 with A = ones.
  // D[m][n] = sum_k B[k][n] + C[m][n]  => every row of D holds the colsums.
  if (tid < 32) {
    const int lane = tid;
    const int col  = lane & 15;
    const int krow = (lane < 16) ? 0 : 2;     // B layout: v0 = K0|K2, v1 = K1|K3
    v2f a; a[0] = 1.0f; a[1] = 1.0f;          // 16x4 all-ones A matrix
    v8f d0 = {0.f, 0.f, 0.f, 0.f, 0.f, 0.f, 0.f, 0.f};
    v8f d1 = {0.f, 0.f, 0.f, 0.f, 0.f, 0.f, 0.f, 0.f};
    for (int r = 0; r < BLOCK; r += 4) {
      v2f bA, bB;
      bA[0] = tile[(r + krow)     * C_PAD + col];
      bA[1] = tile[(r + krow + 1) * C_PAD + col];
      bB[0] = tile[(r + krow)     * C_PAD + 16 + col];
      bB[1] = tile[(r + krow + 1) * C_PAD + 16 + col];
      d0 = __builtin_amdgcn_wmma_f32_16x16x4_f32(false, a, false, bA,
                                                 (short)0, d0, false, false);
      d1 = __builtin_amdgcn_wmma_f32_16x16x4_f32(false, a, false, bB,
                                                 (short)0, d1, false, false);
    }
    // Row 0 of D: VGPR0, lanes 0..15 = classes (colsum)
    if (lane < 16) atomicAdd(&g_psum[b * C_NUM + lane], d0[0]);
    if (lane < 2)  atomicAdd(&g_psum[b * C_NUM + 16 + lane], d1[0]);
  }

  if (tid < C_NUM) {
    atomicAdd(&g_nom[b * C_NUM + tid], s_nom[tid]);
    atomicAdd(&g_cnt[b * C_NUM + tid], s_cnt[tid]);
  }
}

// ---------------------------------------------------------------------------
// Epilogue: O(B*C) log/clamp math -> scalar. Mirrors torch-style BCE clamps.
// ---------------------------------------------------------------------------
__device__ __forceinline__ float neg_log_ratio(float num, float den, bool ok) {
  float r = ok ? num / (den > 0.f ? den : 1.f) : 1.f;
  r = fminf(fmaxf(r, 1e-43f), 1.f);
  float l = logf(r);
  l = fminf(fmaxf(l, -100.f), 0.f);
  return ok ? -l : 0.f;
}

__global__ void semscal_final(const float* __restrict__ g_psum,
                              const float* __restrict__ g_nom,
                              const float* __restrict__ g_cnt,
                              const float* __restrict__ w,
                              float* __restrict__ out, int Bn, int N) {
  __shared__ float per_b[32];
  const int t = threadIdx.x;
  if (t < Bn) {
    float loss_b = 0.f;
    float cnt_b  = 0.f;
    const float Nf = (float)N;
    for (int c = 0; c < C_NUM; ++c) {
      const float psum = g_psum[t * C_NUM + c];
      const float nom  = g_nom [t * C_NUM + c];
      const float ts   = g_cnt [t * C_NUM + c];
      const bool  mask = ts > 0.f;
      const float spec_den = Nf - ts;
      const float spec_nom = (Nf - psum) - (ts - nom);
      float lc = neg_log_ratio(nom, psum, mask && (psum > 0.f))       // precision
               + neg_log_ratio(nom, ts,   mask)                       // recall
               + neg_log_ratio(spec_nom, spec_den,
                               mask && (spec_den > 0.f));             // specificity
      loss_b += lc * w[c];
      cnt_b  += mask ? 1.f : 0.f;
    }
    per_b[t] = loss_b / cnt_b;
  }
  __syncthreads();
  if (t == 0) {
    float s = 0.f;
    for (int bb = 0; bb < Bn; ++bb) s += per_b[bb];
    out[0] = s / (float)Bn;
  }
}

// ---------------------------------------------------------------------------
extern "C" void kernel_launch(void* const* d_in, const int* in_sizes, int n_in,
                              void* d_out, int out_size, void* d_ws, size_t ws_size,
                              hipStream_t stream) {
  const float* pred = (const float*)d_in[0];
  const int*   gt   = (const int*)d_in[1];
  const float* w    = (const float*)d_in[2];
  float*       out  = (float*)d_out;

  const int B = 8;
  const int N = in_sizes[1] / B;          // gt is [B, N]

  float* ws     = (float*)d_ws;
  float* g_psum = ws;
  float* g_nom  = ws + B * C_NUM;
  float* g_cnt  = ws + 2 * B * C_NUM;

  hipMemsetAsync(d_ws, 0, (size_t)(3 * B * C_NUM) * sizeof(float), stream);

  dim3 grid(B * BLOCKS_PER_B);
  semscal_main<<<grid, BLOCK, 0, stream>>>(pred, gt, g_psum, g_nom, g_cnt, N);
  semscal_final<<<1, 32, 0, stream>>>(g_psum, g_nom, g_cnt, w, out, B, N);
}